// FourierOptics2DRenderer_36558761624374
// MI455X (gfx1250) — compile-verified
//
#include <hip/hip_runtime.h>
#include <math.h>

typedef __attribute__((ext_vector_type(2))) float v2f;
typedef __attribute__((ext_vector_type(8))) float v8f;

#define S 64          // pupil size (and output crop size)
#define F 256         // zero-padded FFT size
#define PITCH 66      // LDS row pitch: keeps 8B alignment, kills bank conflicts
#define TWO_PI 6.28318530717958647692f

__global__ void zero_out_kernel(float* out, int n) {
    int i = blockIdx.x * blockDim.x + threadIdx.x;
    if (i < n) out[i] = 0.0f;
}

__launch_bounds__(256)
__global__ void psf_render_kernel(const float* __restrict__ xs,
                                  const float* __restrict__ ys,
                                  const float* __restrict__ zs,
                                  const float* __restrict__ As,
                                  const float* __restrict__ bgs,
                                  const float* __restrict__ ps,
                                  const float* __restrict__ pupil_phase,
                                  float* __restrict__ out) {
    __shared__ float Pre[S * PITCH];           // pupil real  (reused as crop later)
    __shared__ float Pim[S * PITCH];           // pupil imag
    __shared__ float Wre[F * PITCH];           // Wt[v][j] = cos(2pi*(j-32)*v/256)
    __shared__ float Wim[F * PITCH];           // Wt[v][j] = sin(...)
    __shared__ float Tre[F * PITCH];           // T transposed: T_t[v][i]
    __shared__ float Tim[F * PITCH];
    __shared__ float wave_max[8];

    const int e = blockIdx.x;                  // emitter = b*4 + n
    const int b = e >> 2;
    if (ps[e] <= 0.5f) return;                 // masked emitter contributes 0
    const float xe = xs[e], ye = ys[e], ze = zs[e];
    const float Ae = As[e], bge = bgs[e];

    const int t    = threadIdx.x;
    const int lane = t & 31;
    const int wave = t >> 5;
    const int col  = lane & 15;                // fragment M/N index for this lane
    const int hi   = lane >> 4;                // 0: K=0,1  1: K=2,3 in A/B frags

    // ---- twiddle matrix Wt[v][j] = e^{i*2pi*((j-32)*v mod 256)/256} ----
    for (int idx = t; idx < F * S; idx += 256) {
        const int v = idx >> 6, j = idx & 63;
        const int ti = ((j - 32) * v) & 255;   // exact integer angle reduction
        float sn, cs;
        __sincosf(TWO_PI * (float)ti * (1.0f / 256.0f), &sn, &cs);
        Wre[v * PITCH + j] = cs;
        Wim[v * PITCH + j] = sn;
    }

    // ---- pupil: mag * exp(i*(phase0 + prop*z - 2pi*kx*x - 2pi*ky*y)) ----
    for (int idx = t; idx < S * S; idx += 256) {
        const int i = idx >> 6, j = idx & 63;
        const float ui = (2.0f * (float)i / 63.0f - 1.0f) * (1.0f / 0.75f);
        const float vj = (2.0f * (float)j / 63.0f - 1.0f) * (1.0f / 0.75f);
        const float r2 = ui * ui + vj * vj;
        const float mag  = (r2 <= 1.0f) ? 1.0f : 0.0f;
        const float prop = sqrtf(1.0f - fminf(r2, 1.0f));   // sqrt(1-min(R,1)^2)
        const float ph = pupil_phase[idx] + prop * ze
                       - TWO_PI * ((float)(i - 32) * (1.0f / 256.0f)) * xe
                       - TWO_PI * ((float)(j - 32) * (1.0f / 256.0f)) * ye;
        float sn, cs;
        __sincosf(ph, &sn, &cs);
        Pre[i * PITCH + j] = mag * cs;
        Pim[i * PITCH + j] = mag * sn;
    }
    __syncthreads();

    // ---- stage 1: T[i][v] = sum_j P[i][j] * Wt[v][j]   (M=64, N=256, K=64)
    // 64 tiles of 16x16; 8 per wave. 4 independent wmma chains; the Im*Im chain
    // is accumulated positively and SUBTRACTED at writeback (no VALU negation).
    for (int tt = wave * 8; tt < wave * 8 + 8; ++tt) {
        const int ib = (tt >> 4) * 16;
        const int vb = (tt & 15) * 16;
        const int arow = ib + col;
        const int vcol = vb + col;
        v8f aRa = {}, aRb = {}, aIa = {}, aIb = {};
        #pragma unroll
        for (int kb = 0; kb < S; kb += 4) {
            const int k0 = kb + 2 * hi;
            const v2f aR = *(const v2f*)&Pre[arow * PITCH + k0];
            const v2f aI = *(const v2f*)&Pim[arow * PITCH + k0];
            const v2f bR = *(const v2f*)&Wre[vcol * PITCH + k0];
            const v2f bI = *(const v2f*)&Wim[vcol * PITCH + k0];
            aRa = __builtin_amdgcn_wmma_f32_16x16x4_f32(false, aR, false, bR, (short)0, aRa, false, false);
            aRb = __builtin_amdgcn_wmma_f32_16x16x4_f32(false, aI, false, bI, (short)0, aRb, false, false);
            aIa = __builtin_amdgcn_wmma_f32_16x16x4_f32(false, aR, false, bI, (short)0, aIa, false, false);
            aIb = __builtin_amdgcn_wmma_f32_16x16x4_f32(false, aI, false, bR, (short)0, aIb, false, false);
        }
        // C layout: VGPR r holds M=r+8*hi, N=col -> rows consecutive in r: b64 stores
        const int rbase = ib + 8 * hi;
        #pragma unroll
        for (int q = 0; q < 4; ++q) {
            v2f tr, ti;
            tr.x = aRa[2 * q]     - aRb[2 * q];       // Re = ReRe - ImIm
            tr.y = aRa[2 * q + 1] - aRb[2 * q + 1];
            ti.x = aIa[2 * q]     + aIb[2 * q];       // Im = ReIm + ImRe
            ti.y = aIa[2 * q + 1] + aIb[2 * q + 1];
            *(v2f*)&Tre[vcol * PITCH + rbase + 2 * q] = tr;
            *(v2f*)&Tim[vcol * PITCH + rbase + 2 * q] = ti;
        }
    }
    __syncthreads();

    float* crop = Pre;                         // Pre is dead now: reuse as crop tile

    // ---- stage 2: field[u][v] = sum_i Wt[u][i] * T[i][v]   (M=256, N=256, K=64)
    // 256 tiles; 32 per wave. psf = |field|^2; track max; stash crop region.
    float mymax = 0.0f;
    for (int tt = wave * 32; tt < wave * 32 + 32; ++tt) {
        const int ub = (tt >> 4) * 16;
        const int vb = (tt & 15) * 16;
        const int urow = ub + col;
        const int vcol = vb + col;
        v8f aRa = {}, aRb = {}, aIa = {}, aIb = {};
        #pragma unroll
        for (int kb = 0; kb < S; kb += 4) {
            const int k0 = kb + 2 * hi;
            const v2f aR = *(const v2f*)&Wre[urow * PITCH + k0];  // A[u][k]=Wt[u][k]
            const v2f aI = *(const v2f*)&Wim[urow * PITCH + k0];
            const v2f bR = *(const v2f*)&Tre[vcol * PITCH + k0];  // B[k][v]=T_t[v][k]
            const v2f bI = *(const v2f*)&Tim[vcol * PITCH + k0];
            aRa = __builtin_amdgcn_wmma_f32_16x16x4_f32(false, aR, false, bR, (short)0, aRa, false, false);
            aRb = __builtin_amdgcn_wmma_f32_16x16x4_f32(false, aI, false, bI, (short)0, aRb, false, false);
            aIa = __builtin_amdgcn_wmma_f32_16x16x4_f32(false, aR, false, bI, (short)0, aIa, false, false);
            aIb = __builtin_amdgcn_wmma_f32_16x16x4_f32(false, aI, false, bR, (short)0, aIb, false, false);
        }
        const int wv = (vcol + 32) & 255;      // ifftshift+crop map (cols)
        #pragma unroll
        for (int r = 0; r < 8; ++r) {
            const int u  = ub + r + 8 * hi;
            const float re = aRa[r] - aRb[r];
            const float im = aIa[r] + aIb[r];
            const float pw = re * re + im * im;
            mymax = fmaxf(mymax, pw);
            const int wu = (u + 32) & 255;     // ifftshift+crop map (rows)
            if (wu < 64 && wv < 64) crop[wu * PITCH + wv] = pw;
        }
    }

    // ---- block max reduction: wave32 shuffle, then cross-wave via LDS ----
    for (int off = 16; off > 0; off >>= 1)
        mymax = fmaxf(mymax, __shfl_xor(mymax, off, 32));
    if (lane == 0) wave_max[wave] = mymax;
    __syncthreads();
    float mx = wave_max[0];
    for (int w = 1; w < 8; ++w) mx = fmaxf(mx, wave_max[w]);
    const float inv = 1.0f / mx;

    // ---- normalize, scale, accumulate over N with atomics ----
    for (int idx = t; idx < S * S; idx += 256) {
        const int cu = idx >> 6, cv = idx & 63;
        const float val = crop[cu * PITCH + cv] * inv * Ae + bge;
        atomicAdd(&out[b * S * S + idx], val);
    }
}

extern "C" void kernel_launch(void* const* d_in, const int* in_sizes, int n_in,
                              void* d_out, int out_size, void* d_ws, size_t ws_size,
                              hipStream_t stream) {
    const float* x  = (const float*)d_in[0];
    const float* y  = (const float*)d_in[1];
    const float* z  = (const float*)d_in[2];
    const float* A  = (const float*)d_in[3];
    const float* bg = (const float*)d_in[4];
    const float* p  = (const float*)d_in[5];
    const float* ph = (const float*)d_in[6];
    float* out = (float*)d_out;

    const int n_emit = in_sizes[0];            // B*N = 256
    zero_out_kernel<<<(out_size + 255) / 256, 256, 0, stream>>>(out, out_size);
    psf_render_kernel<<<n_emit, 256, 0, stream>>>(x, y, z, A, bg, p, ph, out);
}